// TripletLoss_86363202388634
// MI455X (gfx1250) — compile-verified
//
#include <hip/hip_runtime.h>
#include <hip/hip_bf16.h>
#include <float.h>
#include <cstdint>

// Triplet loss (batch-hard) over an 8192x8192 f32 similarity matrix.
// Memory-bound streaming reduction: 256 MB read once -> ~11 us floor @ 23.3 TB/s.
// CDNA5 paths: global_load_async_to_lds_b128 (+ s_wait_asynccnt) to stage the
// targets vector in LDS, 4x-unrolled b128 non-temporal streaming loads for
// sim_mat (4 loads in flight per wave), wave32 shuffle reductions.

typedef float vf4 __attribute__((ext_vector_type(4)));
typedef int   vi4 __attribute__((ext_vector_type(4)));

// Pointer types matching the async-to-LDS builtin signature (vi4 in as1 / as3).
typedef __attribute__((address_space(1))) vi4* gvi4p;   // global
typedef __attribute__((address_space(3))) vi4* lvi4p;   // LDS (32-bit ptr)

#define THREADS 256
#define MAX_N   8192          // LDS capacity for targets: 32 KB of 320 KB/WGP
#define MARGIN_F 0.1f
#define UNROLL  4

#if defined(__has_builtin)
#if __has_builtin(__builtin_amdgcn_global_load_async_to_lds_b128) && \
    __has_builtin(__builtin_amdgcn_s_wait_asynccnt)
#define TL_USE_ASYNC 1
#endif
#endif

__device__ __forceinline__ float wave_min(float v) {
#pragma unroll
  for (int off = 16; off > 0; off >>= 1) v = fminf(v, __shfl_xor(v, off, 32));
  return v;
}
__device__ __forceinline__ float wave_max(float v) {
#pragma unroll
  for (int off = 16; off > 0; off >>= 1) v = fmaxf(v, __shfl_xor(v, off, 32));
  return v;
}
__device__ __forceinline__ float wave_sum(float v) {
#pragma unroll
  for (int off = 16; off > 0; off >>= 1) v += __shfl_xor(v, off, 32);
  return v;
}

__device__ __forceinline__ void accum4(const vf4 v, const vi4 t, int jbase,
                                       int ti, int ri,
                                       float& pmin, float& nmax) {
#pragma unroll
  for (int c = 0; c < 4; ++c) {
    const float x = v[c];
    if (t[c] == ti) {
      if ((jbase + c) != ri) pmin = fminf(pmin, x);
    } else {
      nmax = fmaxf(nmax, x);
    }
  }
}

__global__ __launch_bounds__(THREADS)
void triplet_rows_kernel(const float* __restrict__ sim,
                         const int*   __restrict__ targets,
                         const int*   __restrict__ idx,
                         float*       __restrict__ row_loss,
                         int n, int ncols) {
  __shared__ int   sT[MAX_N];
  __shared__ float sMin[THREADS / 32];
  __shared__ float sMax[THREADS / 32];

  const int tid  = threadIdx.x;
  const int ccap = (ncols < MAX_N) ? ncols : MAX_N;

  // Stage targets[] into LDS once per (persistent) block: async DMA path.
#ifdef TL_USE_ASYNC
  for (int k = tid * 4; k + 3 < ccap; k += THREADS * 4) {
    __builtin_amdgcn_global_load_async_to_lds_b128(
        (gvi4p)(uintptr_t)(targets + k),
        (lvi4p)(unsigned)(uintptr_t)(&sT[k]),
        0, 0);
  }
  __builtin_amdgcn_s_wait_asynccnt(0);
#else
  for (int k = tid * 4; k + 3 < ccap; k += THREADS * 4)
    *(vi4*)(&sT[k]) = *(const vi4*)(targets + k);
#endif
  __syncthreads();

  const int PASS = THREADS * 4;  // elements covered per block per pass

  for (int row = blockIdx.x; row < n; row += gridDim.x) {
    const int    ti   = sT[row];         // targets[row]
    const int    ri   = idx[row];        // self column to exclude
    const float* rowp = sim + (size_t)ri * (size_t)ncols;

    float pmin =  FLT_MAX;               // hardest positive (min over same-label)
    float nmax = -FLT_MAX;               // hardest negative (max over diff-label)

    // 4 independent b128 NT loads in flight per wave per iteration.
    for (int j = tid * 4; j + 3 < ncols; j += UNROLL * PASS) {
      vf4  v[UNROLL];
      vi4  t[UNROLL];
      int  jj[UNROLL];
      bool ok[UNROLL];
#pragma unroll
      for (int u = 0; u < UNROLL; ++u) {
        jj[u] = j + u * PASS;
        ok[u] = (jj[u] + 3 < ncols);
        if (ok[u]) v[u] = __builtin_nontemporal_load((const vf4*)(rowp + jj[u]));
      }
#pragma unroll
      for (int u = 0; u < UNROLL; ++u)
        if (ok[u]) t[u] = *(const vi4*)(&sT[jj[u]]);
#pragma unroll
      for (int u = 0; u < UNROLL; ++u)
        if (ok[u]) accum4(v[u], t[u], jj[u], ti, ri, pmin, nmax);
    }

    pmin = wave_min(pmin);
    nmax = wave_max(nmax);
    const int wave = tid >> 5, lane = tid & 31;
    if (lane == 0) { sMin[wave] = pmin; sMax[wave] = nmax; }
    __syncthreads();
    if (tid == 0) {
      float pm = sMin[0], nm = sMax[0];
#pragma unroll
      for (int w = 1; w < THREADS / 32; ++w) {
        pm = fminf(pm, sMin[w]);
        nm = fmaxf(nm, sMax[w]);
      }
      const float loss = nm - pm + MARGIN_F;
      row_loss[row] = loss > 0.0f ? loss : 0.0f;
    }
    __syncthreads();  // protect sMin/sMax before next row
  }
}

__global__ __launch_bounds__(THREADS)
void mean_kernel(const float* __restrict__ x, float* __restrict__ out, int n) {
  __shared__ float sS[THREADS / 32];
  float s = 0.0f;
  for (int i = threadIdx.x; i < n; i += THREADS) s += x[i];  // fixed order -> deterministic
  s = wave_sum(s);
  const int wave = threadIdx.x >> 5, lane = threadIdx.x & 31;
  if (lane == 0) sS[wave] = s;
  __syncthreads();
  if (threadIdx.x == 0) {
    float t = 0.0f;
#pragma unroll
    for (int w = 0; w < THREADS / 32; ++w) t += sS[w];
    out[0] = t / (float)n;
  }
}

extern "C" void kernel_launch(void* const* d_in, const int* in_sizes, int n_in,
                              void* d_out, int out_size, void* d_ws, size_t ws_size,
                              hipStream_t stream) {
  const float* sim     = (const float*)d_in[0];
  const int*   targets = (const int*)d_in[1];
  const int*   idx     = (const int*)d_in[2];

  const int n     = in_sizes[1];                       // 8192 rows
  const int ncols = (int)((long long)in_sizes[0] / n); // 8192 cols

  float* row_loss = (float*)d_ws;                      // n floats of scratch

  int nblocks = n < 2048 ? n : 2048;                   // persistent blocks
  hipLaunchKernelGGL(triplet_rows_kernel, dim3(nblocks), dim3(THREADS), 0, stream,
                     sim, targets, idx, row_loss, n, ncols);
  hipLaunchKernelGGL(mean_kernel, dim3(1), dim3(THREADS), 0, stream,
                     row_loss, (float*)d_out, n);
}